// RandomSpatialTransform_2087354106263
// MI455X (gfx1250) — compile-verified
//
#include <hip/hip_runtime.h>
#include <hip/hip_bf16.h>
#include <math.h>

// ---------------------------------------------------------------------------
// RandomSpatialTransform for MI455X (gfx1250)
//   pass 1-3: separable 13-tap Gaussian blur of field (X, Y, Z axes)
//             -> banded GEMM on the WMMA pipe (V_WMMA_F32_16X16X4_F32),
//                inputs staged to LDS with global_load_async_to_lds_b32.
//             The Z pass additionally folds the per-D sum / sum^2 reduction
//             into the same kernel (saves a full 100 MB re-read).
//   pass 4 : final deterministic reduction -> scale[d] = 3 / std_d
//   pass 5 : trilinear warp
// ---------------------------------------------------------------------------

typedef __attribute__((ext_vector_type(2))) float v2f;
typedef __attribute__((ext_vector_type(8))) float v8f;

#define NTOT        2
#define DIM         128
#define CHN         6                         // C*D = 2*3
#define RAD         6
#define KLEN        13
#define TILE_P      (DIM + 2 * RAD)           // 140 padded positions
#define LINES_TOTAL (NTOT * DIM * DIM * CHN)  // 196608 lines per axis
#define LINE_TILES  (LINES_TOTAL / 16)        // 12288 blocks per blur pass
#define FIELD_ELEMS (NTOT * DIM * DIM * DIM * CHN) // 25165824
#define OUT_ELEMS   (NTOT * DIM * DIM * DIM * 2)   // 8388608
#define GROUPS      (FIELD_ELEMS / 3)              // 8388608 per-D elements

// Gaussian kernel, sigma = 2.0, radius 6, normalized (matches reference fp32)
__constant__ float c_w[KLEN] = {
    0.00221820f, 0.00877315f, 0.02702316f, 0.06482519f, 0.12110939f,
    0.17621312f, 0.19967563f, 0.17621312f, 0.12110939f, 0.06482519f,
    0.02702316f, 0.00877315f, 0.00221820f};

// ---------------------------------------------------------------------------
// Blur along one axis. MODE: 0 = Z (stride 6), 1 = Y (stride 768),
// 2 = X (stride 98304). One wave (32 threads) handles 16 lines x 128 pos.
// MODE==0 also emits per-block (sum, sum^2) partials per D component.
// ---------------------------------------------------------------------------
template <int MODE>
__global__ __launch_bounds__(32) void blur_kernel(const float* __restrict__ in,
                                                  float* __restrict__ out,
                                                  float* __restrict__ partials) {
    __shared__ float tile[16 * TILE_P];  // [col][padded pos]
    __shared__ int bases[16];
    __shared__ float red[64];            // 32 sums + 32 sum-of-squares

    const int lane = threadIdx.x;
    constexpr int STRIDE = (MODE == 0) ? CHN : ((MODE == 1) ? DIM * CHN : DIM * DIM * CHN);

    if (lane < 16) {
        int l = blockIdx.x * 16 + lane;
        int base;
        if (MODE == 0) {            // line over Z: l = ((n*128+x)*128+y)*6+ch
            int hi = l / 768;       // n*128+x
            int rem = l - hi * 768; // y*6+ch
            int y = rem / 6;
            base = hi * 98304 + y * 768 + (rem - y * 6);
        } else if (MODE == 1) {     // line over Y: l = ((n*128+x)*128+z)*6+ch
            int hi = l / 768;       // n*128+x
            base = hi * 98304 + (l - hi * 768);  // + z*6+ch
        } else {                    // line over X: l = ((n*128+y)*128+z)*6+ch
            int n = l / 98304;
            base = n * 12582912 + (l - n * 98304);  // + y*768+z*6+ch
        }
        bases[lane] = base;
    }
    __syncthreads();

    // Zero the halo rows (p in [0,6) and [134,140)) with plain DS stores.
    for (int idx = lane; idx < 12 * 16; idx += 32) {
        int j = idx >> 4;
        int col = idx & 15;
        int p = (j < 6) ? j : (j + 128);
        tile[col * TILE_P + p] = 0.0f;
    }

    // Interior: gather 128 x 16 elements straight into LDS (async, no VGPRs).
    unsigned lds_base = (unsigned)(unsigned long long)(&tile[0]);
    for (int idx = lane; idx < DIM * 16; idx += 32) {
        int z = idx >> 4;
        int col = idx & 15;
        const float* gp = in + (long)bases[col] + (long)z * STRIDE;
        unsigned loff = lds_base + (unsigned)((col * TILE_P + z + RAD) * 4);
        asm volatile("global_load_async_to_lds_b32 %0, %1, off"
                     :
                     : "v"(loff), "v"(gp)
                     : "memory");
    }
    asm volatile("s_wait_asynccnt 0" ::: "memory");
    __syncthreads();

    // Banded weight matrix fragments A_c[16x4], c = 0..6 covers K = 0..27.
    // Lane layout (ISA 7.12.2, 32-bit A 16x4): M = lane%16, K = v + 2*(lane>=16).
    const int M = lane & 15;
    const int hi = lane >> 4;
    v2f A[7];
#pragma unroll
    for (int cc = 0; cc < 7; ++cc) {
        int k0 = 4 * cc + 2 * hi;
        int o0 = k0 - M;
        int o1 = k0 + 1 - M;
        A[cc].x = (o0 >= 0 && o0 < KLEN) ? c_w[o0] : 0.0f;
        A[cc].y = (o1 >= 0 && o1 < KLEN) ? c_w[o1] : 0.0f;
    }

    const int col = lane & 15;
    const long obase = bases[col];
    float rs = 0.0f, rq = 0.0f;  // this lane's line-local sum / sum^2

    for (int mt = 0; mt < 8; ++mt) {  // 8 tiles of 16 output positions
        v8f acc = {};
#pragma unroll
        for (int cc = 0; cc < 7; ++cc) {
            // B fragment: N = lane%16, K = v + 2*hi  ->  padded pos p, p+1
            int p = mt * 16 + 4 * cc + 2 * hi;
            v2f b = *(const v2f*)&tile[col * TILE_P + p];
            acc = __builtin_amdgcn_wmma_f32_16x16x4_f32(
                false, A[cc], false, b, (short)0, acc, false, false);
        }
        // D layout: M = v + 8*hi, N = lane%16
#pragma unroll
        for (int v = 0; v < 8; ++v) {
            int z = mt * 16 + v + 8 * hi;
            float val = acc[v];
            out[obase + (long)z * STRIDE] = val;
            if (MODE == 0) { rs += val; rq += val * val; }
        }
    }

    if (MODE == 0) {
        // Fold the 32 per-lane partials in a fixed order (bit-deterministic).
        red[lane] = rs;
        red[32 + lane] = rq;
        __syncthreads();
        if (lane == 0) {
            float S[3] = {0, 0, 0}, Q[3] = {0, 0, 0};
#pragma unroll
            for (int L = 0; L < 32; ++L) {
                int d = (blockIdx.x * 16 + (L & 15)) % 3;  // line % 3 == D comp
                S[d] += red[L];
                Q[d] += red[32 + L];
            }
            float* p = partials + (long)blockIdx.x * 6;
            p[0] = S[0]; p[1] = S[1]; p[2] = S[2];
            p[3] = Q[0]; p[4] = Q[1]; p[5] = Q[2];
        }
    }
}

// ---------------------------------------------------------------------------
// Final reduction over per-block partials -> scale[d] = AMP_SIGMA / std_d
// Fixed iteration/tree order => deterministic across replays.
// ---------------------------------------------------------------------------
__global__ __launch_bounds__(256) void reduce_kernel(const float* __restrict__ partials,
                                                     float* __restrict__ scale) {
    const int tid = threadIdx.x;
    float s[6] = {0, 0, 0, 0, 0, 0};
    for (int i = tid; i < LINE_TILES; i += 256) {
#pragma unroll
        for (int j = 0; j < 6; ++j) s[j] += partials[(long)i * 6 + j];
    }
    __shared__ float sd[6][256];
#pragma unroll
    for (int j = 0; j < 6; ++j) sd[j][tid] = s[j];
    __syncthreads();
    for (int off = 128; off > 0; off >>= 1) {
        if (tid < off) {
#pragma unroll
            for (int j = 0; j < 6; ++j) sd[j][tid] += sd[j][tid + off];
        }
        __syncthreads();
    }
    if (tid == 0) {
        const float Nd = (float)GROUPS;
#pragma unroll
        for (int d = 0; d < 3; ++d) {
            float mean = sd[d][0] / Nd;
            float var = sd[3 + d][0] / Nd - mean * mean;
            scale[d] = 3.0f / sqrtf(var);  // AMP_SIGMA / std
        }
    }
}

// ---------------------------------------------------------------------------
// Trilinear warp: out[n,x,y,z,c] sampled from x[n,:,:,:,c] at grid+disp
// ---------------------------------------------------------------------------
__global__ __launch_bounds__(256) void warp_kernel(const float* __restrict__ xin,
                                                   const float* __restrict__ field,
                                                   const float* __restrict__ scale,
                                                   float* __restrict__ out) {
    const int i = blockIdx.x * 256 + threadIdx.x;  // [0, OUT_ELEMS)
    const float s0 = scale[0], s1 = scale[1], s2 = scale[2];

    const int c = i & 1;
    const int vox = i >> 1;  // ((n*128+x)*128+y)*128+z
    const int z = vox & 127;
    const int y = (vox >> 7) & 127;
    const int xx = (vox >> 14) & 127;
    const int n = vox >> 21;

    const long fb = (long)vox * 6 + c * 3;
    const float dx = field[fb + 0] * s0;
    const float dy = field[fb + 1] * s1;
    const float dz = field[fb + 2] * s2;

    const float lx = fminf(fmaxf((float)xx + dx, 0.0f), 127.0f);
    const float ly = fminf(fmaxf((float)y + dy, 0.0f), 127.0f);
    const float lz = fminf(fmaxf((float)z + dz, 0.0f), 127.0f);
    const float fx = floorf(lx), fy = floorf(ly), fz = floorf(lz);
    const int x0 = (int)fx, y0 = (int)fy, z0 = (int)fz;
    const int x1 = min(x0 + 1, 127), y1 = min(y0 + 1, 127), z1 = min(z0 + 1, 127);
    const float wx = lx - fx, wy = ly - fy, wz = lz - fz;
    const float ux = 1.0f - wx, uy = 1.0f - wy, uz = 1.0f - wz;

    const long nb = (long)n << 21;
    auto V = [&](int a, int b, int cc) -> float {
        return xin[((nb + ((long)a << 14) + (b << 7) + cc) << 1) + c];
    };
    float r = V(x0, y0, z0) * ux * uy * uz + V(x1, y0, z0) * wx * uy * uz +
              V(x0, y1, z0) * ux * wy * uz + V(x0, y0, z1) * ux * uy * wz +
              V(x1, y1, z0) * wx * wy * uz + V(x1, y0, z1) * wx * uy * wz +
              V(x0, y1, z1) * ux * wy * wz + V(x1, y1, z1) * wx * wy * wz;
    out[i] = r;
}

// ---------------------------------------------------------------------------
extern "C" void kernel_launch(void* const* d_in, const int* in_sizes, int n_in,
                              void* d_out, int out_size, void* d_ws, size_t ws_size,
                              hipStream_t stream) {
    (void)in_sizes; (void)n_in; (void)out_size; (void)ws_size;
    const float* x = (const float*)d_in[0];          // [2,128,128,128,2]
    const float* field_rnd = (const float*)d_in[1];  // [2,128,128,128,2,3]
    float* out = (float*)d_out;

    // ws layout: bufA | bufB | partials[12288*6] | scale[3]  (~201.6 MB)
    float* bufA = (float*)d_ws;
    float* bufB = bufA + FIELD_ELEMS;
    float* partials = bufB + FIELD_ELEMS;
    float* scale = partials + (size_t)LINE_TILES * 6;

    const dim3 bgrid(LINE_TILES), bblk(32);
    blur_kernel<2><<<bgrid, bblk, 0, stream>>>(field_rnd, bufA, nullptr);  // X
    blur_kernel<1><<<bgrid, bblk, 0, stream>>>(bufA, bufB, nullptr);       // Y
    blur_kernel<0><<<bgrid, bblk, 0, stream>>>(bufB, bufA, partials);      // Z + stats

    reduce_kernel<<<1, 256, 0, stream>>>(partials, scale);

    warp_kernel<<<OUT_ELEMS / 256, 256, 0, stream>>>(x, bufA, scale, out);
}